// CoupledModel_24670292148945
// MI455X (gfx1250) — compile-verified
//
#include <hip/hip_runtime.h>
#include <hip/hip_bf16.h>

typedef __attribute__((ext_vector_type(2))) float v2f;
typedef __attribute__((ext_vector_type(8))) float v8f;

#define KB_CONST  0.1f
#define NPTS      262144
#define NUM_NETS  24
#define NPAIRS    12
#define WIDTH     8
#define NTILES    (NPTS / 16)

#define LOG2E     1.4426950408889634f
#define LN2       0.6931471805599453f

// Swap the two half-waves (lane L <-> lane L^16) with a single ds_swizzle.
__device__ __forceinline__ float half_swap(float x) {
  return __int_as_float(__builtin_amdgcn_ds_swizzle(__float_as_int(x), 0x401F));
}

// tanh(x) = 1 - 2/(e^{2x}+1), using raw v_exp_f32 / v_rcp_f32.
__device__ __forceinline__ float fast_tanh(float x) {
  float e = __builtin_amdgcn_exp2f(x * (2.0f * LOG2E));   // e^{2x}
  return 1.0f - 2.0f * __builtin_amdgcn_rcpf(e + 1.0f);
}

__device__ __forceinline__ v8f wmma4(v2f a, v2f b, v8f c) {
  // D = A(16x4) * B(4x16) + C, fp32 WMMA
  return __builtin_amdgcn_wmma_f32_16x16x4_f32(false, a, false, b, (short)0, c,
                                               false, false);
}

// Kernel A: all 24 subnets via WMMA, transposed dataflow. Each block handles
// one pair of nets (blockIdx.y); waves grid-stride over 16-point tiles.
__global__ __launch_bounds__(256) void subnet_wmma_kernel(
    const float* __restrict__ x,  const float* __restrict__ W1,
    const float* __restrict__ b1, const float* __restrict__ W2,
    const float* __restrict__ b2, const float* __restrict__ W3,
    float* __restrict__ sub)
{
  const int pair = blockIdx.y;
  const int k0 = 2 * pair, k1 = 2 * pair + 1;
  const int lane = threadIdx.x & 31;
  const int wave = threadIdx.x >> 5;
  const int col  = lane & 15;
  const bool lo  = lane < 16;

  // ---- Stage wave-uniform weights in registers (A-layout operands) ----
  // Layer-1 A matrix (16x4): M = 16 outputs of the net pair, K0/K1 = input
  // weights, K2 = bias row (multiplied by the constant-1 row in B).
  const int netc = (col < 8) ? k0 : k1;
  const int wloc = col & 7;
  v2f a1;
  a1.x = lo ? W1[(netc * WIDTH + wloc) * 2 + 0] : b1[netc * WIDTH + wloc];
  a1.y = lo ? W1[(netc * WIDTH + wloc) * 2 + 1] : 0.0f;

  // Layer-2 A matrix = W2blockT (16x16 block-diagonal over the two nets),
  // as four K-slices of 4, plus a 5th bias slice (K=16 ones row -> b2).
  auto w2blk = [&](int w, int v) -> float {   // W2block[w][v] = W2[k][v][w]
    if (w < 8 && v < 8)   return W2[(k0 * WIDTH + v) * WIDTH + w];
    if (w >= 8 && v >= 8) return W2[(k1 * WIDTH + (v - 8)) * WIDTH + (w - 8)];
    return 0.0f;
  };
  const int khi = lo ? 0 : 2;
  v2f a2[5];
#pragma unroll
  for (int s = 0; s < 4; ++s) {
    a2[s].x = w2blk(4 * s + khi + 0, col);
    a2[s].y = w2blk(4 * s + khi + 1, col);
  }
  a2[4].x = lo ? ((col < 8) ? b2[k0 * WIDTH + col] : b2[k1 * WIDTH + (col - 8)])
               : 0.0f;
  a2[4].y = 0.0f;

  // Layer-3 weights: lane-half selected, so the width-8 dot is 8 plain FMAs.
  float w3v[WIDTH];
#pragma unroll
  for (int i = 0; i < WIDTH; ++i)
    w3v[i] = lo ? W3[k0 * WIDTH + i] : W3[k1 * WIDTH + i];

  const int waveGlobal = blockIdx.x * (blockDim.x >> 5) + wave;
  const int waveStride = gridDim.x * (blockDim.x >> 5);

  for (int t = waveGlobal; t < NTILES; t += waveStride) {
    const int p = t * 16 + col;
    const float2 xv = ((const float2*)x)[p];

    // B operand = xT (4x16): K0=x0, K1=T, K2=1 (bias row), K3=0
    v2f bx;
    bx.x = lo ? xv.x : 1.0f;
    bx.y = lo ? xv.y : 0.0f;

    v8f c0 = {};
    v8f h1 = wmma4(a1, bx, c0);               // h1T in D-layout

    float hv[8], sw[8];
#pragma unroll
    for (int i = 0; i < 8; ++i) hv[i] = fast_tanh(h1[i]);
#pragma unroll
    for (int i = 0; i < 8; ++i) sw[i] = half_swap(hv[i]);

    // Layer 2: 4 K-slices of h1T reshuffled into B-layout + bias slice.
    v8f acc = {};
    v2f bs;
    bs.x = lo ? hv[0] : sw[2];  bs.y = lo ? hv[1] : sw[3];   // K=0..3
    acc = wmma4(a2[0], bs, acc);
    bs.x = lo ? hv[4] : sw[6];  bs.y = lo ? hv[5] : sw[7];   // K=4..7
    acc = wmma4(a2[1], bs, acc);
    bs.x = lo ? sw[0] : hv[2];  bs.y = lo ? sw[1] : hv[3];   // K=8..11
    acc = wmma4(a2[2], bs, acc);
    bs.x = lo ? sw[4] : hv[6];  bs.y = lo ? sw[5] : hv[7];   // K=12..15
    acc = wmma4(a2[3], bs, acc);
    bs.x = lo ? 1.0f : 0.0f;    bs.y = 0.0f;                 // bias ones-row
    acc = wmma4(a2[4], bs, acc);

    // Layer 3: out[p][k] = sum_v tanh(h2T[v][p]) * W3[k][v]
    float s3 = 0.0f;
#pragma unroll
    for (int i = 0; i < 8; ++i) s3 = fmaf(fast_tanh(acc[i]), w3v[i], s3);

    const int k = lo ? k0 : k1;
    sub[p * NUM_NETS + k] = s3;
  }
}

// Kernel B: per-point mixture combine (matches reference math; max-subtracted
// softmax in log2 domain is exactly invariant for w and both weighted sums).
__global__ __launch_bounds__(256) void mixture_kernel(
    const float* __restrict__ x, const float* __restrict__ sub,
    float* __restrict__ out)
{
  const int p = blockIdx.x * blockDim.x + threadIdx.x;
  if (p >= NPTS) return;
  const float T = x[2 * p + 1];
  const float invkT = __builtin_amdgcn_rcpf(KB_CONST * T);

  float vals[NPAIRS], ex[NPAIRS];
  float m = -1e30f;
#pragma unroll
  for (int i = 0; i < NPAIRS; ++i) {
    const float n1  = sub[p * NUM_NETS + 2 * i];
    const float n2r = sub[p * NUM_NETS + 2 * i + 1];
    const float n2  = n2r * n2r;
    const float v   = n1 - T * n2;
    vals[i] = v;
    const float z = n2 * (1.0f / (5.0f * KB_CONST));
    ex[i] = (-v * invkT - z * z) * LOG2E;      // log2-domain exponent
    m = fmaxf(m, ex[i]);
  }
  float S = 0.0f;
#pragma unroll
  for (int i = 0; i < NPAIRS; ++i) {
    ex[i] = __builtin_amdgcn_exp2f(ex[i] - m);
    S += ex[i];
  }
  const float invS = __builtin_amdgcn_rcpf(S);
  float wv = 0.0f, wlp = 0.0f;
#pragma unroll
  for (int i = 0; i < NPAIRS; ++i) {
    const float w = ex[i] * invS;
    wv  = fmaf(w, vals[i], wv);
    wlp = fmaf(w, LN2 * __builtin_amdgcn_logf(w + 1e-9f), wlp);
  }
  out[p] = wv + KB_CONST * T * wlp;
}

extern "C" void kernel_launch(void* const* d_in, const int* in_sizes, int n_in,
                              void* d_out, int out_size, void* d_ws, size_t ws_size,
                              hipStream_t stream) {
  const float* x  = (const float*)d_in[0];
  const float* W1 = (const float*)d_in[1];
  const float* b1 = (const float*)d_in[2];
  const float* W2 = (const float*)d_in[3];
  const float* b2 = (const float*)d_in[4];
  const float* W3 = (const float*)d_in[5];

  float* out = (float*)d_out;        // (N,1) first
  float* sub = out + NPTS;           // then (N,24), concatenated flat

  // 32 chunks x 12 pairs; 8 waves/block -> 256 waves per pair -> 64 tiles/wave
  dim3 gridA(32, NPAIRS);
  subnet_wmma_kernel<<<gridA, 256, 0, stream>>>(x, W1, b1, W2, b2, W3, sub);
  mixture_kernel<<<(NPTS + 255) / 256, 256, 0, stream>>>(x, sub, out);
}